// Block_39814346834309
// MI455X (gfx1250) — compile-verified
//
#include <hip/hip_runtime.h>
#include <math.h>

#define D_MODEL 384
#define D_STATE 16
#define D_CONV  4
#define D_INNER 768
#define DT_RANK 24
#define BATCH   2
#define SEQ     4096
#define NTOK    (BATCH*SEQ)
#define LN_EPS  1e-5f
#define TCHUNK  32

#define KC      32          // K-chunk per LDS tile
#define LDSTR   36          // padded LDS row stride (floats): 144B, 16B-aligned, conflict-free b64 reads

typedef float v2f __attribute__((ext_vector_type(2)));
typedef float v8f __attribute__((ext_vector_type(8)));

__device__ __forceinline__ float silu_f(float x){ return x / (1.0f + __expf(-x)); }
__device__ __forceinline__ float softplus_f(float x){ return (x > 20.0f) ? x : log1pf(__expf(x)); }

#if __has_builtin(__builtin_amdgcn_s_wait_asynccnt)
#define WAIT_ASYNC() __builtin_amdgcn_s_wait_asynccnt(0)
#else
#define WAIT_ASYNC() asm volatile("s_wait_asynccnt 0x0" ::: "memory")
#endif

// async copy 16B from global to LDS (CDNA5 GLOBAL_LOAD_ASYNC_TO_LDS_B128, ASYNCcnt-tracked)
__device__ __forceinline__ void async_g2l_b128(uint32_t lds_addr, const float* gptr){
  uint64_t gaddr = (uint64_t)(uintptr_t)gptr;
  asm volatile("global_load_async_to_lds_b128 %0, %1, off"
               :: "v"(lds_addr), "v"(gaddr) : "memory");
}

// ---------------- LayerNorm: one wave (32 lanes) per 384-wide row ----------------
__global__ void ln_kernel(const float* __restrict__ x, const float* __restrict__ w,
                          const float* __restrict__ b, float* __restrict__ out, int rows){
  int wave = threadIdx.x >> 5;
  int lane = threadIdx.x & 31;
  int row  = blockIdx.x * 8 + wave;
  if (row >= rows) return;
  const float* xr = x + (size_t)row * D_MODEL;
  float v[12];
  float s = 0.f, ss = 0.f;
#pragma unroll
  for (int i = 0; i < 12; ++i){ v[i] = xr[lane + 32*i]; s += v[i]; ss += v[i]*v[i]; }
#pragma unroll
  for (int off = 16; off >= 1; off >>= 1){
    s  += __shfl_xor(s,  off, 32);
    ss += __shfl_xor(ss, off, 32);
  }
  float mu  = s * (1.0f / D_MODEL);
  float var = ss * (1.0f / D_MODEL) - mu * mu;
  float inv = rsqrtf(var + LN_EPS);
  float* orow = out + (size_t)row * D_MODEL;
#pragma unroll
  for (int i = 0; i < 12; ++i){
    int c = lane + 32*i;
    orow[c] = (v[i] - mu) * inv * w[c] + b[c];
  }
}

// ---------------- f32 WMMA GEMM, LDS double-buffered via async-to-LDS ----------------
// C[M,N] = act(A[M,K](lda) @ W[N,K]^T + bias) + res
// 256 threads = 8 waves. Block tile 128(M) x 64(N); wave computes 16x64 (A frag reused 4x).
// ACT: 0=none, 1=relu, 2=softplus
template<int ACT>
__global__ void __launch_bounds__(256)
gemm_wmma(const float* __restrict__ A, const float* __restrict__ W,
          const float* __restrict__ bias, const float* __restrict__ res,
          float* __restrict__ C, int M, int N, int K, int lda, int ldc){
  __shared__ float sA[2][128 * LDSTR];
  __shared__ float sW[2][64 * LDSTR];

  const int tid  = threadIdx.x;
  const int wave = tid >> 5;
  const int lane = tid & 31;
  const int half = lane >> 4;      // K-pair select within a k-step; row-half of C
  const int r    = lane & 15;
  const int m0   = blockIdx.x * 128;
  const int n0   = blockIdx.y * 64;

  const uint32_t sAbase = (uint32_t)(uintptr_t)&sA[0][0];
  const uint32_t sWbase = (uint32_t)(uintptr_t)&sW[0][0];

  // Stage one 128x32 A tile + 64x32 W tile: float4 granules.
  // A: 1024 segs (4/thread). W: 512 segs (2/thread). Past-K segs zero-filled.
  auto issue_tile = [&](int k0, int buf){
#pragma unroll
    for (int i = 0; i < 4; ++i){
      int seg = tid + i*256;
      int row = seg >> 3, kg = seg & 7;
      int k = k0 + kg*4;
      uint32_t la = sAbase + (uint32_t)buf*(128*LDSTR*4) + (uint32_t)(row*LDSTR + kg*4)*4;
      if (k < K) async_g2l_b128(la, A + (size_t)(m0 + row)*lda + k);
      else       *(float4*)(&sA[buf][row*LDSTR + kg*4]) = make_float4(0.f,0.f,0.f,0.f);
    }
#pragma unroll
    for (int i = 0; i < 2; ++i){
      int seg = tid + i*256;
      int row = seg >> 3, kg = seg & 7;
      int k = k0 + kg*4;
      uint32_t la = sWbase + (uint32_t)buf*(64*LDSTR*4) + (uint32_t)(row*LDSTR + kg*4)*4;
      if (k < K) async_g2l_b128(la, W + (size_t)(n0 + row)*K + k);
      else       *(float4*)(&sW[buf][row*LDSTR + kg*4]) = make_float4(0.f,0.f,0.f,0.f);
    }
  };

  v8f acc0 = {}, acc1 = {}, acc2 = {}, acc3 = {};

  auto compute_tile = [&](int buf){
    const float* ab = &sA[buf][(wave*16 + r)*LDSTR + 2*half];
    const float* wb = &sW[buf][r*LDSTR + 2*half];
#pragma unroll
    for (int kk = 0; kk < KC; kk += 4){
      v2f a  = *(const v2f*)(ab + kk);
      v2f b0 = *(const v2f*)(wb + kk);
      v2f b1 = *(const v2f*)(wb + 16*LDSTR + kk);
      v2f b2 = *(const v2f*)(wb + 32*LDSTR + kk);
      v2f b3 = *(const v2f*)(wb + 48*LDSTR + kk);
      acc0 = __builtin_amdgcn_wmma_f32_16x16x4_f32(false, a, false, b0, (short)0, acc0, false, false);
      acc1 = __builtin_amdgcn_wmma_f32_16x16x4_f32(false, a, false, b1, (short)0, acc1, false, false);
      acc2 = __builtin_amdgcn_wmma_f32_16x16x4_f32(false, a, false, b2, (short)0, acc2, false, false);
      acc3 = __builtin_amdgcn_wmma_f32_16x16x4_f32(false, a, false, b3, (short)0, acc3, false, false);
    }
  };

  const int ntiles = (K + KC - 1) / KC;
  issue_tile(0, 0);
  WAIT_ASYNC();
  __syncthreads();
  for (int t = 0; t < ntiles; ++t){
    if (t + 1 < ntiles) issue_tile((t+1)*KC, (t+1)&1);   // overlap with compute of tile t
    compute_tile(t & 1);
    WAIT_ASYNC();
    __syncthreads();
  }

  v8f accs[4] = {acc0, acc1, acc2, acc3};
#pragma unroll
  for (int j = 0; j < 4; ++j){
    int n = n0 + 16*j + r;
    float bv = bias ? bias[n] : 0.0f;
#pragma unroll
    for (int i = 0; i < 8; ++i){
      int m = m0 + wave*16 + 8*half + i;
      float val = accs[j][i] + bv;
      if (ACT == 1) val = fmaxf(val, 0.0f);
      else if (ACT == 2) val = softplus_f(val);
      if (res) val += res[(size_t)m * ldc + n];
      C[(size_t)m * ldc + n] = val;
    }
  }
}

// ---------------- depthwise causal conv(4) + bias + SiLU ----------------
__global__ void conv_silu_kernel(const float* __restrict__ xz, const float* __restrict__ cw,
                                 const float* __restrict__ cb, float* __restrict__ xc){
  int idx = blockIdx.x * blockDim.x + threadIdx.x;   // over NTOK*D_INNER
  if (idx >= NTOK * D_INNER) return;
  int c   = idx % D_INNER;
  int tok = idx / D_INNER;
  int t   = tok % SEQ;
  float s = cb[c];
#pragma unroll
  for (int j = 0; j < D_CONV; ++j){
    int tt = t + j - (D_CONV - 1);
    if (tt >= 0)
      s += cw[c * D_CONV + j] * xz[(size_t)(tok + j - (D_CONV-1)) * (2*D_INNER) + c];
  }
  xc[(size_t)tok * D_INNER + c] = silu_f(s);
}

// ---------------- zero-pad x_proj_w (56 x 768) -> (64 x 768) ----------------
__global__ void pad_xproj_kernel(const float* __restrict__ w, float* __restrict__ wp){
  int idx = blockIdx.x * blockDim.x + threadIdx.x;   // 64*768
  if (idx >= 64 * D_INNER) return;
  int n = idx / D_INNER;
  wp[idx] = (n < (DT_RANK + 2*D_STATE)) ? w[idx] : 0.0f;
}

// ---------------- selective scan (fused skip + gate) ----------------
// grid: BATCH * (D_INNER/128) blocks of 128 threads; one channel per thread.
__global__ void ssm_scan_kernel(const float* __restrict__ xdbl, const float* __restrict__ dt,
                                const float* __restrict__ xc, const float* __restrict__ xz,
                                const float* __restrict__ A_log, const float* __restrict__ Dvec,
                                float* __restrict__ y){
  __shared__ float sB[TCHUNK][D_STATE];
  __shared__ float sC[TCHUNK][D_STATE];
  int b    = blockIdx.x / (D_INNER/128);
  int cblk = blockIdx.x % (D_INNER/128);
  int c    = cblk * 128 + threadIdx.x;

  float Arow[D_STATE];
#pragma unroll
  for (int n = 0; n < D_STATE; ++n) Arow[n] = -__expf(A_log[(size_t)c*D_STATE + n]);
  float Dc = Dvec[c];
  float h[D_STATE];
#pragma unroll
  for (int n = 0; n < D_STATE; ++n) h[n] = 0.0f;

  for (int t0 = 0; t0 < SEQ; t0 += TCHUNK){
    __syncthreads();
    for (int idx = threadIdx.x; idx < TCHUNK*32; idx += blockDim.x){
      int tl = idx >> 5, col = idx & 31;
      float v = xdbl[(size_t)((size_t)b*SEQ + t0 + tl)*64 + DT_RANK + col];
      if (col < D_STATE) sB[tl][col] = v; else sC[tl][col - D_STATE] = v;
    }
    __syncthreads();
    for (int tl = 0; tl < TCHUNK; ++tl){
      size_t tok = (size_t)b*SEQ + t0 + tl;
      float dtv = dt[tok*D_INNER + c];
      float xv  = xc[tok*D_INNER + c];
      float zv  = xz[tok*(2*D_INNER) + D_INNER + c];
      float acc = 0.0f;
#pragma unroll
      for (int n = 0; n < D_STATE; ++n){
        float dA = __expf(dtv * Arow[n]);
        h[n] = dA * h[n] + dtv * sB[tl][n] * xv;
        acc += h[n] * sC[tl][n];
      }
      acc = (acc + xv * Dc) * silu_f(zv);
      y[tok*D_INNER + c] = acc;
    }
  }
}

extern "C" void kernel_launch(void* const* d_in, const int* in_sizes, int n_in,
                              void* d_out, int out_size, void* d_ws, size_t ws_size,
                              hipStream_t stream) {
  const float* x         = (const float*)d_in[0];
  const float* in_proj_w = (const float*)d_in[1];
  const float* conv_w    = (const float*)d_in[2];
  const float* conv_b    = (const float*)d_in[3];
  const float* x_proj_w  = (const float*)d_in[4];
  const float* dt_proj_w = (const float*)d_in[5];
  const float* dt_proj_b = (const float*)d_in[6];
  const float* A_log     = (const float*)d_in[7];
  const float* Dv        = (const float*)d_in[8];
  const float* out_proj_w= (const float*)d_in[9];
  const float* ln1_w     = (const float*)d_in[10];
  const float* ln1_b     = (const float*)d_in[11];
  const float* ln2_w     = (const float*)d_in[12];
  const float* ln2_b     = (const float*)d_in[13];
  const float* ffn_w1    = (const float*)d_in[14];
  const float* ffn_b1    = (const float*)d_in[15];
  const float* ffn_w2    = (const float*)d_in[16];
  const float* ffn_b2    = (const float*)d_in[17];
  float* out = (float*)d_out;

  // workspace carve-up (floats)
  float* ws   = (float*)d_ws;
  float* xn1  = ws;                                  // NTOK*384 (reused as h2)
  float* xz   = xn1  + (size_t)NTOK * D_MODEL;       // NTOK*1536 (reused as ffn1)
  float* xc   = xz   + (size_t)NTOK * 2 * D_INNER;   // NTOK*768
  float* xdbl = xc   + (size_t)NTOK * D_INNER;       // NTOK*64 (padded 56->64)
  float* dtb  = xdbl + (size_t)NTOK * 64;            // NTOK*768
  float* yb   = dtb  + (size_t)NTOK * D_INNER;       // NTOK*768
  float* x2   = yb   + (size_t)NTOK * D_INNER;       // NTOK*384
  float* wpad = x2   + (size_t)NTOK * D_MODEL;       // 64*768

  // 1. LN1
  ln_kernel<<<NTOK/8, 256, 0, stream>>>(x, ln1_w, ln1_b, xn1, NTOK);
  // pad x_proj weights 56 -> 64 rows
  pad_xproj_kernel<<<(64*D_INNER + 255)/256, 256, 0, stream>>>(x_proj_w, wpad);
  // 2. in_proj: (8192x384)x(1536x384)^T -> xz
  gemm_wmma<0><<<dim3(NTOK/128, (2*D_INNER)/64), 256, 0, stream>>>(
      xn1, in_proj_w, nullptr, nullptr, xz, NTOK, 2*D_INNER, D_MODEL, D_MODEL, 2*D_INNER);
  // 3. depthwise conv + SiLU
  conv_silu_kernel<<<(NTOK*D_INNER)/256, 256, 0, stream>>>(xz, conv_w, conv_b, xc);
  // 4. x_proj: (8192x768)x(64x768)^T -> xdbl (cols 0..23 dt, 24..39 B, 40..55 C)
  gemm_wmma<0><<<dim3(NTOK/128, 1), 256, 0, stream>>>(
      xc, wpad, nullptr, nullptr, xdbl, NTOK, 64, D_INNER, D_INNER, 64);
  // 5. dt_proj + softplus: (8192x24 lda64)x(768x24)^T -> dt (K=24 -> zero-padded LDS tail)
  gemm_wmma<2><<<dim3(NTOK/128, D_INNER/64), 256, 0, stream>>>(
      xdbl, dt_proj_w, dt_proj_b, nullptr, dtb, NTOK, D_INNER, DT_RANK, 64, D_INNER);
  // 6. selective scan, fused +x*D skip and *silu(z) gate
  ssm_scan_kernel<<<BATCH*(D_INNER/128), 128, 0, stream>>>(xdbl, dtb, xc, xz, A_log, Dv, yb);
  // 7. out_proj + residual x: -> x2
  gemm_wmma<0><<<dim3(NTOK/128, D_MODEL/64), 256, 0, stream>>>(
      yb, out_proj_w, nullptr, x, x2, NTOK, D_MODEL, D_INNER, D_INNER, D_MODEL);
  // 8. LN2 (h2 reuses xn1)
  ln_kernel<<<NTOK/8, 256, 0, stream>>>(x2, ln2_w, ln2_b, xn1, NTOK);
  // 9. FFN1 + ReLU (ffn1 reuses xz)
  gemm_wmma<1><<<dim3(NTOK/128, (4*D_MODEL)/64), 256, 0, stream>>>(
      xn1, ffn_w1, ffn_b1, nullptr, xz, NTOK, 4*D_MODEL, D_MODEL, D_MODEL, 4*D_MODEL);
  // 10. FFN2 + residual x2 -> out
  gemm_wmma<0><<<dim3(NTOK/128, D_MODEL/64), 256, 0, stream>>>(
      xz, ffn_w2, ffn_b2, x2, out, NTOK, D_MODEL, 4*D_MODEL, 4*D_MODEL, D_MODEL);
}